// PrivateEncoder_26310969655373
// MI455X (gfx1250) — compile-verified
//
#include <hip/hip_runtime.h>

typedef __attribute__((ext_vector_type(2))) float v2f;
typedef __attribute__((ext_vector_type(8))) float v8f;

#define KDIM 768
#define NDIM 768
#define MTOT 16384   // B*P = 64*256
#define BM 128
#define BN 128
#define BK 32
#define LDK (BK + 4)                     // padded LDS row stride (floats)
#define SMEM_BYTES (4 * BM * LDK * 4)    // lA[2][128][36] + lB[2][128][36]
#define BN_EPS 1e-5f

enum { T_NONE = 0, T_BN = 1, T_BNPOS = 2 };
enum { E_RELU = 0, E_STATS = 1, E_SCATTER = 2 };

__device__ __forceinline__ float relu_(float x) { return x > 0.f ? x : 0.f; }

// ---------------- im2col: x[B,3,256,256] -> patches[16384, 768] ----------------
// row = b*256 + hp*16 + wp ; col k = c*256 + kh*16 + kw
__global__ __launch_bounds__(256) void im2col_k(const float* __restrict__ x,
                                                float* __restrict__ patches) {
  int tid = blockIdx.x * 256 + threadIdx.x;   // 16384 * 192 threads (float4 each)
  int k4  = tid % 192;
  int row = tid / 192;
  int k   = k4 * 4;
  int c   = k >> 8;
  int rem = k & 255;
  int kh  = rem >> 4;
  int kw  = rem & 15;
  int b   = row >> 8;
  int p   = row & 255;
  int hp  = p >> 4;
  int wp  = p & 15;
  const float4 v = *(const float4*)(x + (((size_t)(b * 3 + c) * 256 + hp * 16 + kh) * 256 + wp * 16 + kw));
  *(float4*)(patches + (size_t)row * KDIM + k) = v;
}

// ---------------- inverse permutation ----------------
__global__ void invperm_k(const int* __restrict__ perm, int* __restrict__ inv) {
  int b = blockIdx.x, j = threadIdx.x;
  inv[b * 256 + perm[b * 256 + j]] = j;
}

__global__ void zero_k(float* __restrict__ p, int n) {
  int i = blockIdx.x * 256 + threadIdx.x;
  if (i < n) p[i] = 0.f;
}

// ---------------- BN finalize: (sum,sumsq) -> (scale, shift) ----------------
__global__ void bnfin_k(const float* __restrict__ sum, const float* __restrict__ sumsq,
                        const float* __restrict__ gamma, const float* __restrict__ beta,
                        float* __restrict__ sA, float* __restrict__ sC) {
  int n = blockIdx.x * 256 + threadIdx.x;
  if (n >= NDIM) return;
  const float inv = 1.0f / (float)MTOT;
  float mu  = sum[n] * inv;
  float var = fmaxf(sumsq[n] * inv - mu * mu, 0.f);
  float rs  = rsqrtf(var + BN_EPS);
  float g   = gamma[n] * rs;
  sA[n] = g;
  sC[n] = beta[n] - mu * g;
}

// ---------------- fused WMMA GEMM: Out = epilogue(A' @ W^T + bias) ----------------
// A' = transform(A) applied while staging A through registers into LDS.
// W is [N][K] row-major, so B[k][n] = W[n][k]; LDS B stored [n][k].
// Double-buffered LDS: global loads for tile k+1 are issued before computing tile k.
template <int TMODE, int EMODE>
__global__ __launch_bounds__(256)
void gemm_k(const float* __restrict__ A, const float* __restrict__ W,
            const float* __restrict__ bias,
            const float* __restrict__ sA, const float* __restrict__ sC,
            const float* __restrict__ pos,
            float* __restrict__ Out,
            float* __restrict__ sumBuf, float* __restrict__ sumsqBuf,
            const int* __restrict__ invp) {
  extern __shared__ float smem[];
  float* lA = smem;                      // [2][BM][LDK]
  float* lB = smem + 2 * BM * LDK;       // [2][BN][LDK]

  const int tid  = threadIdx.x;
  const int lane = tid & 31;
  const int wave = tid >> 5;
  const int m0 = blockIdx.y * BM;
  const int n0 = blockIdx.x * BN;
  const int wm = (wave >> 2) << 6;   // 0 / 64
  const int wn = (wave & 3) << 5;    // 0 / 32 / 64 / 96

  v8f acc[4][2];
#pragma unroll
  for (int i = 0; i < 4; ++i)
#pragma unroll
    for (int j = 0; j < 2; ++j) acc[i][j] = (v8f){0.f, 0.f, 0.f, 0.f, 0.f, 0.f, 0.f, 0.f};

  const int lrow = tid >> 3;          // 0..31
  const int lcol = (tid & 7) << 2;    // 0..28 (float4 column)

  float4 ra[4], rb[4];                // register staging for the next tile

  auto loadG = [&](int kt) {
#pragma unroll
    for (int i = 0; i < 4; ++i) {
      const int r = lrow + i * 32;
      const int grow = m0 + r;
      float4 v = *(const float4*)(A + (size_t)grow * KDIM + kt + lcol);
      if (TMODE >= T_BN) {
        float4 s = *(const float4*)(sA + kt + lcol);
        float4 c = *(const float4*)(sC + kt + lcol);
        v.x = relu_(v.x * s.x + c.x);
        v.y = relu_(v.y * s.y + c.y);
        v.z = relu_(v.z * s.z + c.z);
        v.w = relu_(v.w * s.w + c.w);
        if (TMODE == T_BNPOS) {
          float4 pp = *(const float4*)(pos + (size_t)(grow & 255) * NDIM + kt + lcol);
          v.x = relu_(v.x + pp.x);
          v.y = relu_(v.y + pp.y);
          v.z = relu_(v.z + pp.z);
          v.w = relu_(v.w + pp.w);
        }
      }
      ra[i] = v;
      rb[i] = *(const float4*)(W + (size_t)(n0 + r) * KDIM + kt + lcol);
    }
  };
  auto storeL = [&](int b) {
#pragma unroll
    for (int i = 0; i < 4; ++i) {
      const int r = lrow + i * 32;
      *(float4*)(&lA[(b * BM + r) * LDK + lcol]) = ra[i];
      *(float4*)(&lB[(b * BM + r) * LDK + lcol]) = rb[i];
    }
  };

  loadG(0);
  storeL(0);
  __syncthreads();

  int buf = 0;
  for (int kt = 0; kt < KDIM; kt += BK) {
    const bool more = (kt + BK) < KDIM;
    if (kt + 2 * BK < KDIM) {  // global_prefetch_b8 two tiles ahead
      __builtin_prefetch(A + (size_t)(m0 + lrow) * KDIM + kt + 2 * BK + lcol, 0, 0);
      __builtin_prefetch(W + (size_t)(n0 + lrow) * KDIM + kt + 2 * BK + lcol, 0, 0);
    }
    if (more) loadG(kt + BK);  // VMEM clause flies while we compute below

    const int lr16 = lane & 15;
#pragma unroll
    for (int kk = 0; kk < BK; kk += 4) {
      const int kl = kk + ((lane >> 4) << 1);  // lanes 0-15: K=k0,k0+1 ; 16-31: k0+2,k0+3
      v2f af[4], bf[2];
#pragma unroll
      for (int mi = 0; mi < 4; ++mi)
        af[mi] = *(const v2f*)(&lA[(buf * BM + wm + mi * 16 + lr16) * LDK + kl]);
#pragma unroll
      for (int ni = 0; ni < 2; ++ni)
        bf[ni] = *(const v2f*)(&lB[(buf * BM + wn + ni * 16 + lr16) * LDK + kl]);
#pragma unroll
      for (int mi = 0; mi < 4; ++mi)
#pragma unroll
        for (int ni = 0; ni < 2; ++ni)
          acc[mi][ni] = __builtin_amdgcn_wmma_f32_16x16x4_f32(
              false, af[mi], false, bf[ni], (short)0, acc[mi][ni], false, false);
    }

    if (more) {
      storeL(buf ^ 1);   // s_wait_loadcnt lands here, after the compute phase
      __syncthreads();   // one barrier per K tile
      buf ^= 1;
    }
  }

  // epilogue: D layout = VGPR v -> rows v (lanes 0-15) / v+8 (lanes 16-31), col = n0+..+lane%16
  const int lr  = lane & 15;
  const int hi8 = (lane >> 4) << 3;
#pragma unroll
  for (int ni = 0; ni < 2; ++ni) {
    const int n = n0 + wn + ni * 16 + lr;
    const float bv = bias[n];
    float s = 0.f, s2 = 0.f;
#pragma unroll
    for (int mi = 0; mi < 4; ++mi) {
#pragma unroll
      for (int v = 0; v < 8; ++v) {
        const int rl = wm + mi * 16 + v + hi8;
        const int grow = m0 + rl;
        const float t = acc[mi][ni][v] + bv;
        if (EMODE == E_RELU) {
          Out[(size_t)grow * NDIM + n] = relu_(t);
        } else if (EMODE == E_STATS) {
          Out[(size_t)grow * NDIM + n] = t;
          s += t;
          s2 += t * t;
        } else {  // E_SCATTER: out row = b*256 + invperm[b][p]
          const int b = grow >> 8;
          const int orow = (b << 8) + invp[grow];
          Out[(size_t)orow * NDIM + n] = t;
        }
      }
    }
    if (EMODE == E_STATS) {
      s  += __shfl_xor(s, 16, 32);   // lane L and L+16 hold the same column
      s2 += __shfl_xor(s2, 16, 32);
      if (lane < 16) {
        atomicAdd(&sumBuf[n], s);
        atomicAdd(&sumsqBuf[n], s2);
      }
    }
  }
}

extern "C" void kernel_launch(void* const* d_in, const int* in_sizes, int n_in,
                              void* d_out, int out_size, void* d_ws, size_t ws_size,
                              hipStream_t stream) {
  const float* x       = (const float*)d_in[0];
  const float* conv0_w = (const float*)d_in[1];   // [768][768] after reshape
  const float* conv0_b = (const float*)d_in[2];
  const float* w1      = (const float*)d_in[3];   // [3][768][768]
  const float* b1      = (const float*)d_in[4];   // [3][768]
  const float* gamma   = (const float*)d_in[5];
  const float* beta    = (const float*)d_in[6];
  const float* pos     = (const float*)d_in[7];   // [256][768]
  const float* mixer_w = (const float*)d_in[8];
  const float* mixer_b = (const float*)d_in[9];
  const int*   perm    = (const int*)d_in[10];    // [64][256]
  float* out = (float*)d_out;

  const size_t SZ  = (size_t)MTOT * KDIM;  // activation buffer elements
  const size_t SZW = (size_t)NDIM * KDIM;  // one 768x768 weight
  float* bufP  = (float*)d_ws;             // patches, then ping-pong
  float* bufA  = bufP + SZ;
  float* stats = bufA + SZ;
  float* sumB  = stats;
  float* sqB   = stats + NDIM;
  float* sAv   = stats + 2 * NDIM;
  float* sCv   = stats + 3 * NDIM;
  int*   invp  = (int*)(stats + 4 * NDIM);

  dim3 gg(NDIM / BN, MTOT / BM);  // (6, 128)

  im2col_k<<<(MTOT * (KDIM / 4)) / 256, 256, 0, stream>>>(x, bufP);
  invperm_k<<<64, 256, 0, stream>>>(perm, invp);

  // patch embed: h0 = ReLU(patches @ W0^T + b0)
  gemm_k<T_NONE, E_RELU><<<gg, 256, SMEM_BYTES, stream>>>(
      bufP, conv0_w, conv0_b, nullptr, nullptr, nullptr, bufA, nullptr, nullptr, nullptr);

  // layer 0: t0 = h0 @ w1[0]^T + b1[0]  (+ batch stats)
  zero_k<<<6, 256, 0, stream>>>(sumB, 2 * NDIM);
  gemm_k<T_NONE, E_STATS><<<gg, 256, SMEM_BYTES, stream>>>(
      bufA, w1, b1, nullptr, nullptr, nullptr, bufP, sumB, sqB, nullptr);
  bnfin_k<<<3, 256, 0, stream>>>(sumB, sqB, gamma, beta, sAv, sCv);

  // layer 1: input = ReLU(BN(t0)) applied on the fly
  zero_k<<<6, 256, 0, stream>>>(sumB, 2 * NDIM);
  gemm_k<T_BN, E_STATS><<<gg, 256, SMEM_BYTES, stream>>>(
      bufP, w1 + SZW, b1 + NDIM, sAv, sCv, nullptr, bufA, sumB, sqB, nullptr);
  bnfin_k<<<3, 256, 0, stream>>>(sumB, sqB, gamma + NDIM, beta + NDIM, sAv, sCv);

  // layer 2
  zero_k<<<6, 256, 0, stream>>>(sumB, 2 * NDIM);
  gemm_k<T_BN, E_STATS><<<gg, 256, SMEM_BYTES, stream>>>(
      bufA, w1 + 2 * SZW, b1 + 2 * NDIM, sAv, sCv, nullptr, bufP, sumB, sqB, nullptr);
  bnfin_k<<<3, 256, 0, stream>>>(sumB, sqB, gamma + 2 * NDIM, beta + 2 * NDIM, sAv, sCv);

  // mixer: input = ReLU(ReLU(BN(t2)) + pos), output scattered through invperm
  gemm_k<T_BNPOS, E_SCATTER><<<gg, 256, SMEM_BYTES, stream>>>(
      bufP, mixer_w, mixer_b, sAv, sCv, pos, out, nullptr, nullptr, invp);
}